// SimpleRNN_84129819394155
// MI455X (gfx1250) — compile-verified
//
#include <hip/hip_runtime.h>
#include <math.h>

// ---- problem dims (fixed by the reference) ----
constexpr int B = 256, T = 512, I = 256, H = 1024, C = 10;
constexpr int KTOT = H + I;          // 1280 concatenated K
constexpr int KC   = 32;             // K chunk
constexpr int NCHUNK = KTOT / KC;    // 40
constexpr int HCHUNK = H / KC;       // 32 chunks come from h_prev/W_hh
constexpr int TM = 32, TN = 64;      // block output tile
constexpr int LDA = 34, LDB = 34;    // padded LDS dword strides (32 data + 2 pad)

typedef __attribute__((ext_vector_type(2))) float    v2f;
typedef __attribute__((ext_vector_type(8))) float    v8f;
typedef __attribute__((ext_vector_type(4))) unsigned v4u;
typedef __attribute__((ext_vector_type(8))) int      v8i;
typedef __attribute__((ext_vector_type(4))) int      v4i;

// Generic (flat) shared-memory pointer -> 32-bit LDS offset for the TDM D#.
__device__ inline unsigned ldsOffset(const void* p) {
    return (unsigned)(unsigned long long)
           (__attribute__((address_space(3))) const char*)p;
}

// Issue one TDM 2-D tile load: rows x KC f32 elements, row stride strideElems,
// into LDS at ldsOff with 2 pad dwords inserted after every 32 dwords
// (matching the LDA/LDB = 34 padded tile layout).
__device__ inline void tdmLoad2D(unsigned ldsOff, const float* g,
                                 unsigned rows, unsigned strideElems) {
    unsigned long long ga = (unsigned long long)g;
    // D# group 0: count=1 | lds_addr | global_addr[56:0] | type=2
    v4u g0 = { 1u,
               ldsOff,
               (unsigned)ga,
               (unsigned)((ga >> 32) & 0x01FFFFFFu) | (2u << 30) };
    // D# group 1:
    //  d0: data_size=2 (4B) | pad_enable | pad_interval=4 (32 dwords) | pad_amount=1 (2 dwords)
    //  d1: tensor_dim0[15:0]=KC << 16
    //  d2: tensor_dim1[15:0]=rows << 16
    //  d3: tile_dim0=KC << 16
    //  d4: tile_dim1=rows  (tile_dim2=0 -> 2-D)
    //  d5: tensor_dim0_stride[31:0]=strideElems ; d6,d7: hi bits / dim1 stride = 0
    v8i g1 = { (int)((2u << 16) | (1u << 20) | (4u << 22) | (1u << 25)),
               (int)((unsigned)KC << 16),
               (int)(rows << 16),
               (int)((unsigned)KC << 16),
               (int)rows,
               (int)strideElems,
               0, 0 };
    v4i z4 = { 0, 0, 0, 0 };
    v8i z8 = { 0, 0, 0, 0, 0, 0, 0, 0 };
    __builtin_amdgcn_tensor_load_to_lds(g0, g1, z4, z4, z8, 0);
}

// One recurrence step:  h_out = tanh( [h_in | x_t] @ [W_hh | W_ih]^T + b_ih + b_hh )
__global__ __launch_bounds__(128) void rnn_step(
    const float* __restrict__ x,     // [B,T,I]
    const float* __restrict__ W_ih,  // [H,I]
    const float* __restrict__ W_hh,  // [H,H]
    const float* __restrict__ b_ih,  // [H]
    const float* __restrict__ b_hh,  // [H]
    const float* __restrict__ h_in,  // [B,H]
    float* __restrict__ h_out,       // [B,H]
    int t, int kcBegin)
{
    __shared__ float As[2][TM * LDA];   // A tile: [m][k] (padded rows)
    __shared__ float Bs[2][TN * LDB];   // B tile: rows of W (= W^T cols): [n][k]

    const int tid  = threadIdx.x;        // 0..127
    const int lane = tid & 31;
    const int w    = tid >> 5;           // wave 0..3
    const int mw   = w >> 1;             // 0..1 : 16-row slab
    const int nw   = w & 1;              // 0..1 : 32-col slab
    const int bm0  = blockIdx.y * TM;    // batch-row tile base
    const int bn0  = blockIdx.x * TN;    // output-col tile base

    // ---- TDM tile issue: wave 0 -> A tile, wave 1 -> B tile (EXEC-independent,
    // SGPR descriptors; all inputs wave-uniform) ----
    auto issueChunk = [&](int kc, int buf) {
        const int k0 = kc * KC;
        if (w == 0) {
            const float* aSrc; unsigned aStride;
            if (kc < HCHUNK) { aSrc = h_in + (long)bm0 * H + k0;                aStride = H; }
            else             { aSrc = x + ((long)bm0 * T + t) * I + (k0 - H);   aStride = T * I; }
            tdmLoad2D(ldsOffset(&As[buf][0]), aSrc, TM, aStride);
        } else { // w == 1
            const float* bSrc; unsigned bStride;
            if (kc < HCHUNK) { bSrc = W_hh + (long)bn0 * H + k0;                bStride = H; }
            else             { bSrc = W_ih + (long)bn0 * I + (k0 - H);          bStride = I; }
            tdmLoad2D(ldsOffset(&Bs[buf][0]), bSrc, TN, bStride);
        }
    };

    v8f acc0 = {};   // C tile (nw*32 + 0..15)
    v8f acc1 = {};   // C tile (nw*32 + 16..31)

    // f32 WMMA fragment coordinates (ISA 7.12.2):
    //   A 16x4: lanes 0-15 -> M=lane, K={0,1}; lanes 16-31 -> M=lane-16, K={2,3}
    //   B 4x16: lanes 0-15 -> N=lane, K={0,1}; lanes 16-31 -> N=lane-16, K={2,3}
    const int fl  = lane & 15;
    const int kHi = (lane >> 4) << 1;       // 0 or 2

    if (w < 2) {
        issueChunk(kcBegin, 0);
        __builtin_amdgcn_s_wait_tensorcnt(0);
    }
    __syncthreads();

    int buf = 0;
    for (int kc = kcBegin; kc < NCHUNK; ++kc) {
        if (kc + 1 < NCHUNK && w < 2) issueChunk(kc + 1, buf ^ 1);

        const float* aBase  = &As[buf][(mw * 16 + fl) * LDA];
        const float* bBase0 = &Bs[buf][(nw * 32 + fl) * LDB];
        const float* bBase1 = bBase0 + 16 * LDB;
        #pragma unroll
        for (int kk = 0; kk < KC; kk += 4) {
            v2f a  = *(const v2f*)(aBase  + kk + kHi);
            v2f b0 = *(const v2f*)(bBase0 + kk + kHi);
            v2f b1 = *(const v2f*)(bBase1 + kk + kHi);
            acc0 = __builtin_amdgcn_wmma_f32_16x16x4_f32(
                       false, a, false, b0, (short)0, acc0, false, false);
            acc1 = __builtin_amdgcn_wmma_f32_16x16x4_f32(
                       false, a, false, b1, (short)0, acc1, false, false);
        }

        if (w < 2) __builtin_amdgcn_s_wait_tensorcnt(0);
        __syncthreads();
        buf ^= 1;
    }

    // ---- epilogue: + bias, tanh, store ----
    // C/D layout: VGPR r, lanes 0-15 -> M=r ; lanes 16-31 -> M=8+r ; N=lane&15
    const int rowBase = bm0 + mw * 16 + ((lane >> 4) << 3);
    const int colBase = bn0 + nw * 32 + fl;
    #pragma unroll
    for (int r = 0; r < 8; ++r) {
        int row = rowBase + r;
        {
            int col = colBase;
            float y = acc0[r] + b_ih[col] + b_hh[col];
            h_out[(long)row * H + col] = tanhf(y);
        }
        {
            int col = colBase + 16;
            float y = acc1[r] + b_ih[col] + b_hh[col];
            h_out[(long)row * H + col] = tanhf(y);
        }
    }
}

// out[b,c] = h[b,:] . W_fc[c,:] + b_fc[c]    (tiny: 256x10x1024)
__global__ __launch_bounds__(128) void rnn_fc(
    const float* __restrict__ h, const float* __restrict__ W_fc,
    const float* __restrict__ b_fc, float* __restrict__ out)
{
    __shared__ float red[128];
    const int b = blockIdx.x, tid = threadIdx.x;
    const float* hb = h + (long)b * H;
    for (int c = 0; c < C; ++c) {
        float s = 0.f;
        const float* wc = W_fc + (long)c * H;
        for (int k = tid; k < H; k += 128) s += hb[k] * wc[k];
        red[tid] = s;
        __syncthreads();
        #pragma unroll
        for (int off = 64; off > 0; off >>= 1) {
            if (tid < off) red[tid] += red[tid + off];
            __syncthreads();
        }
        if (tid == 0) out[(long)b * C + c] = red[0] + b_fc[c];
        __syncthreads();
    }
}

extern "C" void kernel_launch(void* const* d_in, const int* in_sizes, int n_in,
                              void* d_out, int out_size, void* d_ws, size_t ws_size,
                              hipStream_t stream)
{
    (void)in_sizes; (void)n_in; (void)out_size; (void)ws_size;
    const float* x    = (const float*)d_in[0];
    const float* W_ih = (const float*)d_in[1];
    const float* W_hh = (const float*)d_in[2];
    const float* b_ih = (const float*)d_in[3];
    const float* b_hh = (const float*)d_in[4];
    const float* W_fc = (const float*)d_in[5];
    const float* b_fc = (const float*)d_in[6];
    float* out = (float*)d_out;

    // workspace: two [B,H] hidden-state buffers (2 MB)
    float* h0 = (float*)d_ws;
    float* h1 = h0 + (long)B * H;

    dim3 grid(H / TN, B / TM);   // 16 x 8 = 128 workgroups
    dim3 block(128);

    for (int t = 0; t < T; ++t) {
        const float* hin = (t & 1) ? h1 : h0;  // unused (and unread) at t==0
        float*       hout = (t & 1) ? h0 : h1;
        int kcBegin = (t == 0) ? HCHUNK : 0;   // h_0 == 0: skip recurrent K-chunks
        rnn_step<<<grid, block, 0, stream>>>(x, W_ih, W_hh, b_ih, b_hh,
                                             hin, hout, t, kcBegin);
    }
    // t=511 (odd) wrote h0
    rnn_fc<<<dim3(B), dim3(128), 0, stream>>>(h0, W_fc, b_fc, out);
}